// SpectralNetLoss_4286377361973
// MI455X (gfx1250) — compile-verified
//
#include <hip/hip_runtime.h>
#include <math.h>

typedef __attribute__((ext_vector_type(2))) float v2f;
typedef __attribute__((ext_vector_type(8))) float v8f;

#define MDIM 8192
#define KDIM 64
#define WAVES 8      // waves per block (8 * 32 = 256 threads)
#define JCHUNK 16    // gridDim.x ; each wave handles (512/JCHUNK)=32 j-tiles
#define TJ (512 / JCHUNK)

// ---------------- Kernel 1: D[i] = sum_j W[i][j] ----------------
__global__ __launch_bounds__(256) void rowsum_kernel(const float* __restrict__ W,
                                                     float* __restrict__ D) {
    __shared__ float red[256];
    const int row = blockIdx.x;
    const float* wr = W + (size_t)row * MDIM;
    float s = 0.f;
    for (int c = threadIdx.x; c < MDIM; c += 256)
        s += __builtin_nontemporal_load(wr + c);
    red[threadIdx.x] = s;
    __syncthreads();
    for (int off = 128; off > 0; off >>= 1) {
        if (threadIdx.x < off) red[threadIdx.x] += red[threadIdx.x + off];
        __syncthreads();
    }
    if (threadIdx.x == 0) D[row] = red[0];
}

// ---------------- Kernel 2: rinv[i]=1/sqrt(D[i]); sq[i]=|Yn_i|^2 ----------------
__global__ __launch_bounds__(256) void norm_kernel(const float* __restrict__ Y,
                                                   const float* __restrict__ D,
                                                   float* __restrict__ rinv,
                                                   float* __restrict__ sq) {
    const int i = blockIdx.x * blockDim.x + threadIdx.x;
    if (i >= MDIM) return;
    const float r = 1.0f / sqrtf(D[i]);
    const float4* y = (const float4*)(Y + (size_t)i * KDIM);
    float s = 0.f;
#pragma unroll
    for (int k = 0; k < KDIM / 4; ++k) {
        float4 v = y[k];
        float a = v.x * r, b = v.y * r, c = v.z * r, d = v.w * r;
        s += a * a + b * b + c * c + d * d;
    }
    rinv[i] = r;
    sq[i] = s;
}

// ---------------- Kernel 3: fused Gram (fp32 WMMA) + clamp + W-weighted sum ----------------
__global__ __launch_bounds__(256) void tile_kernel(const float* __restrict__ W,
                                                   const float* __restrict__ Y,
                                                   const float* __restrict__ rinv,
                                                   const float* __restrict__ sq,
                                                   float* __restrict__ partials) {
    __shared__ float wred[WAVES];
    const int lane = threadIdx.x & 31;
    const int wave = threadIdx.x >> 5;
    const int lrow = lane & 15;   // row within tile (A) / col within tile (B, C, D)
    const int lhi  = lane >> 4;   // half-wave select

    const int it = blockIdx.y * WAVES + wave;   // i-tile index [0,512)
    const int i0 = it * 16;

    // --- A strip: Yn rows i0..i0+15, all K, WMMA 16x4 fp32 layout ---
    // lane L: row i0+(L&15); step kk covers K = 4*kk + 2*(L>>4) + {0,1}
    const int arow = i0 + lrow;
    const float ri = rinv[arow];
    v2f a[16];
    {
        const float* ybase = Y + (size_t)arow * KDIM + 2 * lhi;
#pragma unroll
        for (int kk = 0; kk < 16; ++kk) {
            v2f t = *(const v2f*)(ybase + 4 * kk);
            a[kk] = t * ri;
        }
    }
    // sq_i for the 8 C-rows this lane owns (m = 8*lhi + v)
    float sqi[8];
    {
        const int ibase = i0 + 8 * lhi;
#pragma unroll
        for (int v = 0; v < 8; ++v) sqi[v] = sq[ibase + v];
    }
    const size_t wrow = (size_t)(i0 + 8 * lhi) * MDIM;

    float acc = 0.f;
    const int jt0 = blockIdx.x * TJ;
    for (int jt = jt0; jt < jt0 + TJ; ++jt) {
        const int j0 = jt * 16;
        // --- B strip (= Yn^T tile): same per-lane pattern with j-base ---
        const int brow = j0 + lrow;
        const float rj = rinv[brow];
        const float* ybase = Y + (size_t)brow * KDIM + 2 * lhi;

        v8f c = {};
#pragma unroll
        for (int kk = 0; kk < 16; ++kk) {
            v2f bt = *(const v2f*)(ybase + 4 * kk);
            bt = bt * rj;
            // D = A x B + C, full fp32 (V_WMMA_F32_16X16X4_F32)
            c = __builtin_amdgcn_wmma_f32_16x16x4_f32(
                    false, a[kk], false, bt, (short)0, c, false, false);
        }

        // --- fuse: acc += W_ij * max(sq_i + sq_j - 2*G_ij, 0) ---
        const float sqj = sq[j0 + lrow];                       // n = lane&15
        const float* wp = W + wrow + j0 + lrow;
        float lacc = 0.f;
#pragma unroll
        for (int v = 0; v < 8; ++v) {
            float w = __builtin_nontemporal_load(wp + (size_t)v * MDIM);
            float d = sqi[v] + sqj - 2.0f * c[v];
            d = fmaxf(d, 0.0f);
            lacc = fmaf(w, d, lacc);
        }
        acc += lacc;
    }

    // --- deterministic reduction: lanes -> wave, waves -> block partial ---
#pragma unroll
    for (int off = 16; off > 0; off >>= 1)
        acc += __shfl_xor(acc, off, 32);
    if (lane == 0) wred[wave] = acc;
    __syncthreads();
    if (threadIdx.x == 0) {
        float s = 0.f;
#pragma unroll
        for (int w = 0; w < WAVES; ++w) s += wred[w];
        partials[blockIdx.y * gridDim.x + blockIdx.x] = s;
    }
}

// ---------------- Kernel 4: fixed-order final reduce + scale ----------------
__global__ __launch_bounds__(256) void final_kernel(const float* __restrict__ partials,
                                                    int n, float* __restrict__ out) {
    __shared__ float red[256];
    float s = 0.f;
    for (int i = threadIdx.x; i < n; i += 256) s += partials[i];
    red[threadIdx.x] = s;
    __syncthreads();
    for (int off = 128; off > 0; off >>= 1) {
        if (threadIdx.x < off) red[threadIdx.x] += red[threadIdx.x + off];
        __syncthreads();
    }
    if (threadIdx.x == 0) out[0] = red[0] / (2.0f * (float)MDIM);
}

extern "C" void kernel_launch(void* const* d_in, const int* in_sizes, int n_in,
                              void* d_out, int out_size, void* d_ws, size_t ws_size,
                              hipStream_t stream) {
    (void)in_sizes; (void)n_in; (void)out_size; (void)ws_size;
    const float* W = (const float*)d_in[0];
    const float* Y = (const float*)d_in[1];

    float* ws       = (float*)d_ws;
    float* D        = ws;               // 8192 floats
    float* rinv     = ws + MDIM;        // 8192 floats
    float* sq       = ws + 2 * MDIM;    // 8192 floats
    float* partials = ws + 3 * MDIM;    // JCHUNK * 64 = 1024 floats

    rowsum_kernel<<<MDIM, 256, 0, stream>>>(W, D);
    norm_kernel<<<MDIM / 256, 256, 0, stream>>>(Y, D, rinv, sq);

    dim3 grid(JCHUNK, (MDIM / 16) / WAVES);   // (16, 64) = 1024 blocks, 8192 waves
    tile_kernel<<<grid, 256, 0, stream>>>(W, Y, rinv, sq, partials);

    final_kernel<<<1, 256, 0, stream>>>(partials, JCHUNK * ((MDIM / 16) / WAVES),
                                        (float*)d_out);
}